// LossSupervisedTags_83880711290948
// MI455X (gfx1250) — compile-verified
//
#include <hip/hip_runtime.h>

// ---------------------------------------------------------------------------
// LossSupervisedTags — MI455X (gfx1250, wave32)
//
// Bandwidth-bound masked-SSE reduction (~185 MB HBM traffic -> ~8 us at
// 23.3 TB/s). Strategy:
//  * preds det-half (142 MB, read-once) streamed with NT temporal hint
//    (global_load_b128 th:TH_LOAD_NT) so it does not evict L2-resident data
//  * heatmaps (36 MB) + masks (2 MB) loaded RT: reused x4 across S, served
//    from the 192 MB L2 after first touch
//  * deterministic fixed-order tree reductions (no float atomics)
//  * final 32-lane reduction on the matrix pipe: two chained
//    V_WMMA_F32_16X16X4_F32 with an all-ones B matrix
// ---------------------------------------------------------------------------

typedef float v2f __attribute__((ext_vector_type(2)));
typedef float v4f __attribute__((ext_vector_type(4)));
typedef float v8f __attribute__((ext_vector_type(8)));

#define BDIM     256
#define GRID_DET 4096

// Problem shape (from reference setup_inputs)
#define B_   32
#define S_   4
#define NP   17           // N_PARTS
#define CC   34           // (1+TAG_DIM)*N_PARTS
#define HW   16384        // 128*128
#define HW4  4096         // HW/4
#define MM   30
#define N_DET (B_*S_*NP*HW)   // 35,651,584 det elements
#define NC4   (N_DET/4)       // 8,912,896 float4 chunks

// Deterministic in-block tree reduction over BDIM floats in LDS.
__device__ __forceinline__ float block_reduce(float v, float* red) {
  const int tid = threadIdx.x;
  red[tid] = v;
  __syncthreads();
  for (int s = BDIM / 2; s > 0; s >>= 1) {
    if (tid < s) red[tid] += red[tid + s];
    __syncthreads();
  }
  return red[0];
}

// ---- Kernel 1: det loss partial sums ---------------------------------------
// sum over (b,s,p,h,w) of masks[b,h,w] * (preds[b,s,p,h,w] - heatmaps[b,p,h,w])^2
__global__ __launch_bounds__(BDIM) void det_partial(
    const float* __restrict__ preds,
    const float* __restrict__ masks,
    const float* __restrict__ heatmaps,
    float* __restrict__ ws_det) {
  __shared__ float red[BDIM];
  const v4f* __restrict__ preds4 = reinterpret_cast<const v4f*>(preds);
  const v4f* __restrict__ heat4  = reinterpret_cast<const v4f*>(heatmaps);
  const v4f* __restrict__ mask4  = reinterpret_cast<const v4f*>(masks);

  const int stride = GRID_DET * BDIM;
  float acc = 0.0f;
  for (int c = blockIdx.x * BDIM + threadIdx.x; c < NC4; c += stride) {
    const int hw4   = c & (HW4 - 1);     // chunk within one H*W plane
    const int plane = c >> 12;           // plane index over (B*S*NP)
    const int bs    = plane / NP;        // (b*S + s)
    const int p     = plane - bs * NP;   // part channel
    const int b     = bs >> 2;           // S == 4

    const int pidx = (bs * CC + p) * HW4 + hw4;
    const int hidx = (b * NP + p) * HW4 + hw4;
    const int midx = b * HW4 + hw4;

    // Read-once stream: non-temporal so it doesn't evict heatmaps/masks
    // from L2 (lowered as global_load_b128 th:TH_LOAD_NT).
    const v4f d = __builtin_nontemporal_load(preds4 + pidx);
    // Reused operands: default RT policy, L2-resident after first touch.
    const v4f h = heat4[hidx];
    const v4f m = mask4[midx];

    float t;
    t = d.x - h.x; acc = fmaf(m.x, t * t, acc);
    t = d.y - h.y; acc = fmaf(m.y, t * t, acc);
    t = d.z - h.z; acc = fmaf(m.z, t * t, acc);
    t = d.w - h.w; acc = fmaf(m.w, t * t, acc);
  }

  const float s = block_reduce(acc, red);
  if (threadIdx.x == 0) ws_det[blockIdx.x] = s;
}

// ---- Kernel 2: tag loss partial sums (one block per (b,s)) ------------------
// pred_tag = preds[b, s, 17 + 0, :].flat[idx]; loss = (pt - gt)^2 * vis
__global__ __launch_bounds__(BDIM) void tag_partial(
    const float* __restrict__ preds,
    const long long* __restrict__ kp_idx,   // int64 (B, M, P)
    const float* __restrict__ kp_vis,       // (B, M, P)
    const float* __restrict__ gt_tags,      // (B, 1, M, P)
    float* __restrict__ ws_tag) {
  __shared__ float red[BDIM];
  const int bs = blockIdx.x;           // b*S + s
  const int b  = bs >> 2;              // S == 4
  const float* __restrict__ tagbase =
      preds + ((size_t)bs * CC + NP) * HW;   // start of tag channels for (b,s)

  float acc = 0.0f;
  for (int i = threadIdx.x; i < MM * NP; i += BDIM) {
    const int m = i / NP;
    const int p = i - m * NP;
    const size_t kidx = ((size_t)b * MM + m) * NP + p;
    const long long idx = kp_idx[kidx];          // in [0, NP*HW)
    const float vis = kp_vis[kidx];
    const float gt  = gt_tags[kidx];             // tag_dim == 1
    // Read-once random gather from the 142 MB tag half: non-temporal.
    const float pt  = __builtin_nontemporal_load(tagbase + idx);
    const float d   = pt - gt;
    acc = fmaf(d * d, vis, acc);
  }

  const float s = block_reduce(acc, red);
  if (threadIdx.x == 0) ws_tag[bs] = s;
}

// ---- Kernel 3: finalize -----------------------------------------------------
// Deterministic fixed-order sum of the scaled partials; the last 32->1
// cross-lane step runs on the matrix pipe: two chained v_wmma_f32_16x16x4_f32
// with an all-ones B matrix. A-layout (ISA 7.12.2, 16x4 f32): VGPR0 holds
// K=0 (lanes 0-15, M=lane) and K=2 (lanes 16-31, M=lane-16); VGPR1 (K=1/K=3)
// is zeroed. With B=ones: D[m][n] = A[m][0] + A[m][2], i.e. lane m+16 folds
// into lane m, replicated across all N. Summing the 8 D VGPRs per lane gives
// the half-sums H0 (lanes 0-15) / H1 (lanes 16-31); a second identical WMMA
// yields H0+H1 = total in every lane. All waves execute the WMMAs with full
// EXEC (only the store is predicated), per the EXEC-all-ones restriction.
__global__ __launch_bounds__(BDIM) void finalize(
    const float* __restrict__ ws_det,
    const float* __restrict__ ws_tag,
    float* __restrict__ out) {
  __shared__ float red[BDIM];
  const int tid = threadIdx.x;

  const float c_det = 1.0f / (float)N_DET;        // HM_W * mean over B,S,17,H,W
  const float c_tag = 0.001f / (float)(B_ * S_);  // TAG_W * mean over B,S

  float acc = 0.0f;
  for (int i = tid; i < GRID_DET; i += BDIM) acc += ws_det[i] * c_det;
  for (int i = tid; i < B_ * S_;  i += BDIM) acc += ws_tag[i] * c_tag;

  red[tid] = acc;
  __syncthreads();

  // Every wave redundantly reduces the same 256 staged values (keeps EXEC
  // all-ones in every wave for the WMMAs below).
  const int lane = tid & 31;
  float v = 0.0f;
  #pragma unroll
  for (int j = 0; j < 8; ++j) v += red[lane + 32 * j];

  v2f a;  a.x = v;    a.y = 0.0f;   // K=0 / K=2 slots carry data, K=1/K=3 zero
  v2f bo; bo.x = 1.0f; bo.y = 1.0f; // B = ones(4x16)
  v8f c0 = {};
  v8f d1 = __builtin_amdgcn_wmma_f32_16x16x4_f32(
      false, a, false, bo, (short)0, c0, false, false);

  float t = d1[0] + d1[1] + d1[2] + d1[3] + d1[4] + d1[5] + d1[6] + d1[7];

  v2f a2; a2.x = t; a2.y = 0.0f;
  v8f c1 = {};
  v8f d2 = __builtin_amdgcn_wmma_f32_16x16x4_f32(
      false, a2, false, bo, (short)0, c1, false, false);

  if (tid == 0) out[0] = d2[0];
}

// ---------------------------------------------------------------------------
extern "C" void kernel_launch(void* const* d_in, const int* in_sizes, int n_in,
                              void* d_out, int out_size, void* d_ws, size_t ws_size,
                              hipStream_t stream) {
  const float*     preds  = (const float*)d_in[0];
  const float*     masks  = (const float*)d_in[1];
  const long long* kp_idx = (const long long*)d_in[2];
  const float*     kp_vis = (const float*)d_in[3];
  const float*     gt     = (const float*)d_in[4];
  const float*     hm     = (const float*)d_in[5];

  float* ws_det = (float*)d_ws;            // GRID_DET partials
  float* ws_tag = ws_det + GRID_DET;       // B_*S_ partials

  det_partial<<<GRID_DET, BDIM, 0, stream>>>(preds, masks, hm, ws_det);
  tag_partial<<<B_ * S_, BDIM, 0, stream>>>(preds, kp_idx, kp_vis, gt, ws_tag);
  finalize<<<1, BDIM, 0, stream>>>(ws_det, ws_tag, (float*)d_out);
}